// EdgeConv_22479858828031
// MI455X (gfx1250) — compile-verified
//
#include <hip/hip_runtime.h>
#include <math.h>

// EdgeConv fused kernel for MI455X (gfx1250, wave32):
//   y = LeakyReLU( BN( W @ x ) ) ; out = max over K
// GEMM via V_WMMA_F32_16X16X4_F32 (full fp32, matches reference precision).
// HBM-bound (AI ~61 flop/B): x (336 MB) is streamed exactly once.
// X tile staged into LDS by the Tensor Data Mover (TENSOR_LOAD_TO_LDS, TENSORcnt),
// falling back to GLOBAL_LOAD_ASYNC_TO_LDS_B128 (ASYNCcnt) if TDM is unavailable.

typedef float v2f __attribute__((ext_vector_type(2)));
typedef float v4f __attribute__((ext_vector_type(4)));
typedef float v8f __attribute__((ext_vector_type(8)));
typedef int   v4i __attribute__((ext_vector_type(4)));
typedef int   v8i __attribute__((ext_vector_type(8)));
typedef unsigned int v4u __attribute__((ext_vector_type(4)));

#define AS1 __attribute__((address_space(1)))
#define AS3 __attribute__((address_space(3)))

#define BN_EPS 1e-5f
#define SLOPE  0.2f

constexpr int CIN   = 128;
constexpr int COUT  = 128;
constexpr int NPTS  = 1024;
constexpr int KNB   = 20;            // neighbor axis
constexpr int NB    = 4;             // n-values per workgroup
constexpr int MCOL  = NB * KNB;      // 80 columns per workgroup
constexpr int NTILE = MCOL / 16;     // 5 WMMA column tiles
constexpr int CSTEP = CIN / 4;       // 32 k-steps of 4 channels (WMMA K=4)
constexpr int ROWSTRIDE = NPTS * KNB;    // 20480 floats between channels in x

__global__ __launch_bounds__(256)
void edgeconv_wmma_kernel(const float* __restrict__ x,
                          const float* __restrict__ Wm,
                          const float* __restrict__ gamma,
                          const float* __restrict__ beta,
                          const float* __restrict__ rmean,
                          const float* __restrict__ rvar,
                          float* __restrict__ out)
{
    __shared__ float xs[CIN * MCOL];   // 40 KB: X tile, later reused as y tile
    __shared__ float s_scale[COUT];
    __shared__ float s_shift[COUT];

    const int tid  = threadIdx.x;
    const int lane = tid & 31;
    const int wave = tid >> 5;                 // 0..7 -> C_out rows [16w, 16w+16)
    const int blk  = blockIdx.x;
    const int b    = blk >> 8;                 // / (NPTS/NB) == /256
    const int n0   = (blk & 255) * NB;

    // ---- fold BN into per-channel scale/shift (inference form) ----
    if (tid < COUT) {
        float inv = rsqrtf(rvar[tid] + BN_EPS);
        float sc  = gamma[tid] * inv;
        s_scale[tid] = sc;
        s_shift[tid] = beta[tid] - rmean[tid] * sc;
    }

    // ---- stage X tile (CIN x MCOL f32 = 40 KB) into LDS ----
    // tile row c (80 contiguous floats) starts at x[(b*CIN+c)*ROWSTRIDE + n0*KNB]
    const float* xbase = x + (size_t)b * CIN * ROWSTRIDE + n0 * KNB;

#if __has_builtin(__builtin_amdgcn_tensor_load_to_lds) && \
    __has_builtin(__builtin_amdgcn_s_wait_tensorcnt)
    // --- Tensor Data Mover: one 2D-tile descriptor per workgroup ---
    // D# (ISA 8.3/8.4): 2D tensor, data_size=4B, tile 80x128, row stride 20480 elems.
    if (wave == 0) {                     // TDM ignores EXEC: issue from one wave only
        const unsigned lds_off = (unsigned)(size_t)(AS3 float*)&xs[0];
        const unsigned long long ga = (unsigned long long)(size_t)xbase;
        v4u g0;
        g0.x = 1u;                                        // count=1 (valid), user mode
        g0.y = lds_off;                                   // lds_addr (bytes)
        g0.z = (unsigned)(ga & 0xFFFFFFFFu);              // global_addr[31:0]
        g0.w = (unsigned)((ga >> 32) & 0x01FFFFFFu)       // global_addr[56:32]
             | (2u << 30);                                // type = 2 ("image")
        v8i g1;
        g1[0] = (2 << 16);                // workgroup_mask=0 | data_size=2 (4B)
        g1[1] = (MCOL << 16);             // abar_addr=0 | tensor_dim0[15:0]=80
        g1[2] = (CIN << 16);              // tensor_dim0[31:16]=0 | tensor_dim1[15:0]=128
        g1[3] = (MCOL << 16);             // tensor_dim1[31:16]=0 | tile_dim0=80
        g1[4] = CIN;                      // tile_dim1=128 | tile_dim2=0
        g1[5] = ROWSTRIDE;                // tensor_dim0_stride[31:0]=20480
        g1[6] = 0;                        // stride[47:32]=0 | dim1_stride[15:0]=0
        g1[7] = 0;                        // dim1_stride[47:16]=0
        v4i gz4 = {0, 0, 0, 0};                    // groups 2/3 unused (2D tensor)
        v8i gz8 = {0, 0, 0, 0, 0, 0, 0, 0};        // trailing group unused
        __builtin_amdgcn_tensor_load_to_lds(g0, g1, gz4, gz4, gz8, /*cpol=*/0);
        __builtin_amdgcn_s_wait_tensorcnt(0);
    }
#elif __has_builtin(__builtin_amdgcn_global_load_async_to_lds_b128)
    // --- async global->LDS (ASYNCcnt), 10 b128 issues per thread ---
    {
        const int c  = tid >> 1;                   // 0..127
        const int q0 = (tid & 1) * 10;             // first float4 of this half-row
        const float* gsrc = xbase + (size_t)c * ROWSTRIDE + q0 * 4;
        float*       ldst = &xs[c * MCOL + q0 * 4];
#pragma unroll
        for (int j = 0; j < 10; ++j) {
            __builtin_amdgcn_global_load_async_to_lds_b128(
                (AS1 v4i*)(gsrc + j * 4), (AS3 v4i*)(ldst + j * 4), 0, 0);
        }
#if __has_builtin(__builtin_amdgcn_s_wait_asynccnt)
        __builtin_amdgcn_s_wait_asynccnt(0);
#else
        asm volatile("s_wait_asynccnt 0" ::: "memory");
#endif
    }
#else
    // --- fallback: global -> VGPR -> LDS (non-temporal stream) ---
    {
        const int c  = tid >> 1;
        const int q0 = (tid & 1) * 10;
        const float* gsrc = xbase + (size_t)c * ROWSTRIDE + q0 * 4;
        float*       ldst = &xs[c * MCOL + q0 * 4];
#pragma unroll
        for (int j = 0; j < 10; ++j) {
            v4f v = __builtin_nontemporal_load((const v4f*)(gsrc + j * 4));
            *(v4f*)(ldst + j * 4) = v;
        }
    }
#endif

    // ---- preload this wave's A fragments: W rows [16w,16w+16), all C_in ----
    // A(16x4) f32 layout: lane L -> M = L%16 ; VGPR0 holds K = 2*(L/16), VGPR1 K+1
    const int arow = wave * 16 + (lane & 15);
    const int cb   = (lane >> 4) * 2;
    const float* wrow = Wm + arow * CIN + cb;
    v2f a[CSTEP];
#pragma unroll
    for (int s = 0; s < CSTEP; ++s)
        a[s] = *(const v2f*)(wrow + 4 * s);

    __syncthreads();   // X tile visible to all waves (TDM/async awaited above)

    // ---- GEMM: 5 output tiles (16 C_out x 16 cols), K-reduce over C_in ----
    // B(4x16) f32 layout: lane L -> N = L%16 ; VGPR0 holds K = 2*(L/16), VGPR1 K+1
    v8f acc[NTILE] = {};
    const int mcol = lane & 15;
#pragma unroll
    for (int s = 0; s < CSTEP; ++s) {
        const int c = 4 * s + cb;
#pragma unroll
        for (int t = 0; t < NTILE; ++t) {
            v2f bf;
            bf.x = xs[c       * MCOL + t * 16 + mcol];
            bf.y = xs[(c + 1) * MCOL + t * 16 + mcol];
            acc[t] = __builtin_amdgcn_wmma_f32_16x16x4_f32(
                false, a[s], false, bf, (short)0, acc[t], false, false);
        }
    }

    __syncthreads();   // all waves done reading xs -> safe to reuse as y tile

    // ---- epilogue: BN + LeakyReLU, scatter to LDS as y[c_out][m] ----
    // C/D 16x16 f32 layout: VGPR v, lane L -> M = v + 8*(L/16), N = L%16
    {
        const int rbase = wave * 16 + 8 * (lane >> 4);
#pragma unroll
        for (int v = 0; v < 8; ++v) {
            const int co = rbase + v;
            const float sc = s_scale[co];
            const float sh = s_shift[co];
#pragma unroll
            for (int t = 0; t < NTILE; ++t) {
                float y = acc[t][v] * sc + sh;
                y = (y >= 0.f) ? y : SLOPE * y;
                xs[co * MCOL + t * 16 + mcol] = y;
            }
        }
    }
    // wave w wrote rows [16w,16w+16) and only reads those rows below;
    // DS ops are in-order within a wave -> no extra barrier needed.

    // ---- max over K=20 per (c_out, n); 16 rows x 4 n per wave ----
    const int r = lane & 15;
#pragma unroll
    for (int j = 0; j < 2; ++j) {
        const int nl = (lane >> 4) + 2 * j;                  // 0..3
        const float* yrow = &xs[(wave * 16 + r) * MCOL + nl * KNB];
        float m = yrow[0];
#pragma unroll
        for (int k = 1; k < KNB; ++k) m = fmaxf(m, yrow[k]);
        out[((size_t)b * COUT + wave * 16 + r) * NPTS + n0 + nl] = m;
    }
}

extern "C" void kernel_launch(void* const* d_in, const int* in_sizes, int n_in,
                              void* d_out, int out_size, void* d_ws, size_t ws_size,
                              hipStream_t stream) {
    const float* x     = (const float*)d_in[0];
    const float* Wm    = (const float*)d_in[1];
    const float* gamma = (const float*)d_in[2];
    const float* beta  = (const float*)d_in[3];
    const float* rmean = (const float*)d_in[4];
    const float* rvar  = (const float*)d_in[5];
    float* out = (float*)d_out;

    // 32 batches * 256 n-blocks = 8192 workgroups, 256 threads (8 wave32) each
    edgeconv_wmma_kernel<<<dim3(32 * (NPTS / NB)), dim3(256), 0, stream>>>(
        x, Wm, gamma, beta, rmean, rvar, out);
}